// HyperConv_9302899163737
// MI455X (gfx1250) — compile-verified
//
#include <hip/hip_runtime.h>
#include <hip/hip_bf16.h>
#include <math.h>

#define N_ITEMS 200000
#define EMB 100
#define BATCH 100
#define HIST_MAX 50
#define LAYERS 3
#define NNZ_A 6400000
#define NNZ_H 25000

#define R_ROWS (BATCH * HIST_MAX)          // 5000
#define R_TILES ((R_ROWS + 15) / 16)       // 313
#define R_PAD (R_TILES * 16)               // 5008 (padded so WMMA loads never guard)

typedef float v2f __attribute__((ext_vector_type(2)));
typedef float v8f __attribute__((ext_vector_type(8)));

// ---------------------------------------------------------------------------
// COO SpMM scatter: y[row[k],:] += val[k] * x[col[k],:]
// One wave32 per nonzero; lanes stride the 100-wide row: gather and atomic
// scatter are both 128B/wave coalesced. f32 relaxed agent-scope atomics ->
// global_atomic_add_f32 resolving in L2 (80MB dense operand fits in 192MB L2).
// EMB=100 unrolled: 3 full strides + 4-lane tail.
// ---------------------------------------------------------------------------
__global__ void spmm_atomic_kernel(const int* __restrict__ row,
                                   const int* __restrict__ col,
                                   const float* __restrict__ val,
                                   const float* __restrict__ x,
                                   float* __restrict__ y, int nnz) {
    int wave = (int)((blockIdx.x * blockDim.x + threadIdx.x) >> 5);
    int lane = (int)(threadIdx.x & 31);
    if (wave >= nnz) return;
    int r = row[wave];
    int c = col[wave];
    float v = val[wave];
    const float* xr = x + (long long)c * EMB;
    float* yr = y + (long long)r * EMB;

    (void)__hip_atomic_fetch_add(&yr[lane],      v * xr[lane],
                                 __ATOMIC_RELAXED, __HIP_MEMORY_SCOPE_AGENT);
    (void)__hip_atomic_fetch_add(&yr[lane + 32], v * xr[lane + 32],
                                 __ATOMIC_RELAXED, __HIP_MEMORY_SCOPE_AGENT);
    (void)__hip_atomic_fetch_add(&yr[lane + 64], v * xr[lane + 64],
                                 __ATOMIC_RELAXED, __HIP_MEMORY_SCOPE_AGENT);
    if (lane < EMB - 96)
        (void)__hip_atomic_fetch_add(&yr[lane + 96], v * xr[lane + 96],
                                     __ATOMIC_RELAXED, __HIP_MEMORY_SCOPE_AGENT);
}

// ---------------------------------------------------------------------------
// float4-vectorized: buf *= gamma[layer+1]; acc += buf
// (layer 0 seeds acc = emb*gamma[0] + buf). n4 = n/4 elements.
// ---------------------------------------------------------------------------
__global__ void scale_acc_kernel(float4* __restrict__ buf,
                                 float4* __restrict__ acc,
                                 const float4* __restrict__ emb0,
                                 const float* __restrict__ gamma,
                                 int layer, int n4) {
    int i = (int)(blockIdx.x * blockDim.x + threadIdx.x);
    if (i >= n4) return;
    float g = gamma[layer + 1];
    float4 t = buf[i];
    t.x *= g; t.y *= g; t.z *= g; t.w *= g;
    buf[i] = t;
    float4 a;
    if (layer == 0) {
        float g0 = gamma[0];
        float4 e = emb0[i];
        a.x = e.x * g0 + t.x; a.y = e.y * g0 + t.y;
        a.z = e.z * g0 + t.z; a.w = e.w * g0 + t.w;
    } else {
        a = acc[i];
        a.x += t.x; a.y += t.y; a.z += t.z; a.w += t.w;
    }
    acc[i] = a;
}

// ---------------------------------------------------------------------------
// P[r] = dot(hist[r, :], W1[EMB:2*EMB]) via V_WMMA_F32_16X16X4_F32.
// One wave per 16 rows; hist is padded to R_PAD rows so A loads are
// unconditional global_load_b64. B holds the weight pair (lane-half uniform,
// broadcast load) multiplied by a column-0 mask -> v_cndmask, no exec
// divergence around the WMMA (EXEC stays all-ones as required).
// A layout (ISA 7.12.2): lane L = row M=L&15, K = 2*(L>>4)+{0,1} in 2 VGPRs;
// B mirrors with N across lanes. D column 0 -> lanes 0 (M=0..7) / 16 (M=8..15).
// ---------------------------------------------------------------------------
__global__ void alpha_wmma_kernel(const float* __restrict__ hist,
                                  const float* __restrict__ W1,
                                  float* __restrict__ P, int nTiles) {
    int wave = (int)((blockIdx.x * blockDim.x + threadIdx.x) >> 5);
    int lane = (int)(threadIdx.x & 31);
    if (wave >= nTiles) return;   // wave-uniform exit
    int row0 = wave * 16;

    int m = lane & 15;
    int khalf = (lane >> 4) * 2;              // 0 / 2
    float mask = (m == 0) ? 1.0f : 0.0f;      // keep only B column 0
    const float* hrow = hist + (long long)(row0 + m) * EMB + khalf;
    const float* wp = W1 + EMB + khalf;

    v8f c = {};
#pragma unroll
    for (int k0 = 0; k0 < EMB; k0 += 4) {
        v2f a  = *(const v2f*)(hrow + k0);    // 8B-aligned: (100r+k0+khalf)*4
        v2f wv = *(const v2f*)(wp + k0);
        v2f b;
        b.x = wv.x * mask;
        b.y = wv.y * mask;
        c = __builtin_amdgcn_wmma_f32_16x16x4_f32(
                false, a, false, b, (short)0, c, false, false);
    }

    if (m == 0) {                 // lanes 0 and 16 hold D column 0
        int mbase = (lane >> 4) * 8;
#pragma unroll
        for (int j = 0; j < 8; ++j)
            P[row0 + mbase + j] = c[j];   // P padded to R_PAD: no guard
    }
}

// ---------------------------------------------------------------------------
// Per-batch attention: logits = ue.W1[:E] + P[b,:] + mask(-99999), softmax
// over HIST_MAX, weighted sum of hist rows / denom. One block per batch elem.
// ---------------------------------------------------------------------------
__global__ void attention_out_kernel(const float* __restrict__ hist,
                                     const float* __restrict__ P,
                                     const float* __restrict__ ue_tab,
                                     const int* __restrict__ user,
                                     const int* __restrict__ hist_len,
                                     const float* __restrict__ W1,
                                     float* __restrict__ out) {
    int b = (int)blockIdx.x;
    int tid = (int)threadIdx.x;
    __shared__ float red[128];
    __shared__ float alpha[HIST_MAX];
    __shared__ float su_s;

    int u = user[b];
    float p = 0.0f;
    for (int e = tid; e < EMB; e += 128)
        p += ue_tab[(long long)u * EMB + e] * W1[e];
    red[tid] = p;
    __syncthreads();
    for (int s = 64; s > 0; s >>= 1) {
        if (tid < s) red[tid] += red[tid + s];
        __syncthreads();
    }
    if (tid == 0) su_s = red[0];
    __syncthreads();

    int len = hist_len[b];
    if (tid < HIST_MAX) {
        float a = su_s + P[b * HIST_MAX + tid];
        if (tid >= len) a += -99999.0f;   // matches reference mask exactly
        alpha[tid] = a;
    }
    __syncthreads();

    if (tid == 0) {  // 50-wide softmax: cheap serial pass
        float mx = -3.0e38f;
        for (int h = 0; h < HIST_MAX; ++h) mx = fmaxf(mx, alpha[h]);
        float s = 0.0f;
        for (int h = 0; h < HIST_MAX; ++h) {
            float e = expf(alpha[h] - mx);
            alpha[h] = e;
            s += e;
        }
        float inv = 1.0f / s;
        for (int h = 0; h < HIST_MAX; ++h) alpha[h] *= inv;
    }
    __syncthreads();

    float denom = (len == 0) ? 1.0f : (float)len;
    for (int e = tid; e < EMB; e += 128) {
        float s = 0.0f;
        for (int h = 0; h < HIST_MAX; ++h)
            s += alpha[h] * hist[((long long)b * HIST_MAX + h) * EMB + e];
        out[b * EMB + e] = s / denom;
    }
}

extern "C" void kernel_launch(void* const* d_in, const int* in_sizes, int n_in,
                              void* d_out, int out_size, void* d_ws, size_t ws_size,
                              hipStream_t stream) {
    (void)in_sizes; (void)n_in; (void)out_size; (void)ws_size;

    const int*   adj_row  = (const int*)d_in[0];
    const int*   adj_col  = (const int*)d_in[1];
    const float* adj_val  = (const float*)d_in[2];
    const int*   hist_row = (const int*)d_in[3];
    const int*   hist_col = (const int*)d_in[4];
    const float* hist_val = (const float*)d_in[5];
    const int*   hist_len = (const int*)d_in[6];
    const int*   user     = (const int*)d_in[7];
    const float* embedding      = (const float*)d_in[8];
    const float* user_embedding = (const float*)d_in[9];
    const float* gamma_k        = (const float*)d_in[10];
    const float* W_1            = (const float*)d_in[11];

    float* acc      = (float*)d_out;                       // item_embeddings
    float* user_out = acc + (size_t)N_ITEMS * EMB;         // [BATCH, EMB]

    float* buf0    = (float*)d_ws;
    float* buf1    = buf0 + (size_t)N_ITEMS * EMB;
    float* histbuf = buf1 + (size_t)N_ITEMS * EMB;         // R_PAD x EMB
    float* P       = histbuf + (size_t)R_PAD * EMB;        // R_PAD

    const size_t itemBytes = (size_t)N_ITEMS * EMB * sizeof(float);
    const int nElem = N_ITEMS * EMB;

    dim3 blk(256);
    const int wavesPerBlock = 256 / 32;
    const int spmmGridA = (NNZ_A + wavesPerBlock - 1) / wavesPerBlock;

    // --- 3-layer gamma-weighted hypergraph propagation ---
    const float* cur = embedding;
    float* bufs[2] = { buf0, buf1 };
    for (int i = 0; i < LAYERS; ++i) {
        float* nxt = bufs[i & 1];
        hipMemsetAsync(nxt, 0, itemBytes, stream);
        spmm_atomic_kernel<<<spmmGridA, blk, 0, stream>>>(
            adj_row, adj_col, adj_val, cur, nxt, NNZ_A);
        const int n4 = nElem / 4;
        scale_acc_kernel<<<(n4 + 255) / 256, blk, 0, stream>>>(
            (float4*)nxt, (float4*)acc, (const float4*)embedding,
            gamma_k, i, n4);
        cur = nxt;
    }

    // --- history aggregation SpMM (tiny; padded rows zeroed for WMMA) ---
    hipMemsetAsync(histbuf, 0, (size_t)R_PAD * EMB * sizeof(float), stream);
    const int spmmGridH = (NNZ_H + wavesPerBlock - 1) / wavesPerBlock;
    spmm_atomic_kernel<<<spmmGridH, blk, 0, stream>>>(
        hist_row, hist_col, hist_val, acc, histbuf, NNZ_H);

    // --- attention logits via f32 WMMA, then softmax + weighted sum ---
    const int alphaThreads = R_TILES * 32;
    alpha_wmma_kernel<<<(alphaThreads + 127) / 128, dim3(128), 0, stream>>>(
        histbuf, W_1, P, R_TILES);

    attention_out_kernel<<<BATCH, dim3(128), 0, stream>>>(
        histbuf, P, user_embedding, user, hist_len, W_1, user_out);
}